// LSTM_36679020708055
// MI455X (gfx1250) — compile-verified
//
#include <hip/hip_runtime.h>
#include <hip/hip_bf16.h>

typedef float v2f __attribute__((ext_vector_type(2)));
typedef float v8f __attribute__((ext_vector_type(8)));

// ---------------------------------------------------------------------------
// Phase 1: xproj[T,40] = x[T,30] @ W_ih^T + (b_ih + b_hh), via f32 WMMA.
// One wave per 16-row (timestep) tile. N padded 40->48 (3 tiles of 16),
// K padded 30->32 (8 k-steps of 4). V_WMMA_F32_16X16X4_F32.
// ---------------------------------------------------------------------------
__global__ __launch_bounds__(256) void xproj_wmma(
    const float* __restrict__ x, const float* __restrict__ W_ih,
    const float* __restrict__ b_ih, const float* __restrict__ b_hh,
    float* __restrict__ xproj, int Mtiles)
{
    const int lane = threadIdx.x & 31;
    const int wid  = blockIdx.x * (blockDim.x >> 5) + (threadIdx.x >> 5);
    if (wid >= Mtiles) return;                 // wave-uniform exit

    const int m  = lane & 15;                  // M (and N) index within tile
    const int kh = lane >> 4;                  // lane half selects K pair

    // A fragments: lanes 0-15 -> K = 4ks+0,1 ; lanes 16-31 -> K = 4ks+2,3
    v2f A[8];
#pragma unroll
    for (int ks = 0; ks < 8; ++ks) {
        const int k   = ks * 4 + kh * 2;
        const int row = wid * 16 + m;
        float a0 = (k     < 30) ? x[row * 30 + k    ] : 0.f;
        float a1 = (k + 1 < 30) ? x[row * 30 + k + 1] : 0.f;
        A[ks].x = a0; A[ks].y = a1;
    }

#pragma unroll
    for (int nt = 0; nt < 3; ++nt) {
        const int  n   = nt * 16 + m;
        const bool nok = (n < 40);
        const float bias = nok ? (b_ih[n] + b_hh[n]) : 0.f;

        v8f acc = {0.f, 0.f, 0.f, 0.f, 0.f, 0.f, 0.f, 0.f};
#pragma unroll
        for (int ks = 0; ks < 8; ++ks) {
            const int k = ks * 4 + kh * 2;
            v2f B;
            // B[k][n] = W_ih^T[k][n] = W_ih[n*30 + k]
            B.x = (nok && (k     < 30)) ? W_ih[n * 30 + k    ] : 0.f;
            B.y = (nok && (k + 1 < 30)) ? W_ih[n * 30 + k + 1] : 0.f;
            acc = __builtin_amdgcn_wmma_f32_16x16x4_f32(
                false, A[ks], false, B, (short)0, acc, false, false);
        }

        if (nok) {
#pragma unroll
            for (int r = 0; r < 8; ++r) {
                const int orow = wid * 16 + r + kh * 8;  // C/D: vgpr r -> M=r (+8 for hi half)
                xproj[orow * 40 + n] = acc[r] + bias;
            }
        }
    }
}

// ---------------------------------------------------------------------------
// Phase 2+3: sequential LSTM scan, single wave32, wave-synchronous.
// Lane mapping: lanes 0..9 -> gates i (rows 0..9) and o (rows 30..39),
// lanes 10..19 -> f, lanes 20..29 -> g(tanh). h broadcast via LDS.
// Depth-4 register prefetch queue on xproj hides L2 latency.
// Output projection folded in via shfl_xor reduction.
// ---------------------------------------------------------------------------
__global__ __launch_bounds__(32) void lstm_seq(
    const float* __restrict__ xproj, const float* __restrict__ W_hh,
    const float* __restrict__ W_out, const float* __restrict__ b_out,
    float* __restrict__ out, int T)
{
    __shared__ __align__(16) float hsh[16];
    const int l = threadIdx.x;

    const int r1 = (l < 30) ? l : 0;           // primary gate row
    const int r2 = (l < 10) ? (30 + l) : 0;    // o-gate row (lanes 0..9)

    float w1[10], w2[10];
#pragma unroll
    for (int k = 0; k < 10; ++k) {
        w1[k] = W_hh[r1 * 10 + k];
        w2[k] = W_hh[r2 * 10 + k];
    }
    const float wo = (l < 10) ? W_out[l] : 0.f;
    const float bo = b_out[0];
    const bool  isG = (l >= 20) && (l < 30);   // tanh gate lanes

    float c = 0.f;
    if (l < 16) hsh[l] = 0.f;
    __syncthreads();

    // prefetch queue, depth 4
    float xqa[4], xqb[4];
#pragma unroll
    for (int s = 0; s < 4; ++s) {
        xqa[s] = xproj[s * 40 + r1];
        xqb[s] = (l < 10) ? xproj[s * 40 + 30 + l] : 0.f;
    }

    for (int tb = 0; tb < T; tb += 4) {
#pragma unroll
        for (int s = 0; s < 4; ++s) {
            const int t = tb + s;
            const float xp1 = xqa[s];
            const float xp2 = xqb[s];
            const int tn = t + 4;
            if (tn < T) {                       // uniform guard; issue next loads early
                xqa[s] = xproj[tn * 40 + r1];
                xqb[s] = (l < 10) ? xproj[tn * 40 + 30 + l] : 0.f;
            }

            // broadcast-read h from LDS (same address in all lanes)
            float h[10];
            {
                const float4 hA = *reinterpret_cast<const float4*>(&hsh[0]);
                const float4 hB = *reinterpret_cast<const float4*>(&hsh[4]);
                const float2 hC = *reinterpret_cast<const float2*>(&hsh[8]);
                h[0] = hA.x; h[1] = hA.y; h[2] = hA.z; h[3] = hA.w;
                h[4] = hB.x; h[5] = hB.y; h[6] = hB.z; h[7] = hB.w;
                h[8] = hC.x; h[9] = hC.y;
            }

            // two-chain dot products to halve dependency depth
            float ea = 0.f, oa = 0.f, eb = 0.f, ob = 0.f;
#pragma unroll
            for (int k = 0; k < 10; k += 2) {
                ea = fmaf(w1[k],     h[k],     ea);
                oa = fmaf(w1[k + 1], h[k + 1], oa);
                eb = fmaf(w2[k],     h[k],     eb);
                ob = fmaf(w2[k + 1], h[k + 1], ob);
            }
            const float acc1 = xp1 + (ea + oa);
            const float acc2 = xp2 + (eb + ob);

            // activations: sigmoid = 1/(1+e^-x); tanh = 2/(1+e^-2x) - 1
            const float z1   = isG ? (-2.f * acc1) : (-acc1);
            const float rc1  = __builtin_amdgcn_rcpf(1.f + __expf(z1));
            const float act1 = isG ? fmaf(2.f, rc1, -1.f) : rc1;
            const float act2 = __builtin_amdgcn_rcpf(1.f + __expf(-acc2)); // o gate

            // exchange f and g into lanes 0..9
            const float fg = __shfl(act1, l + 10, 32);
            const float gg = __shfl(act1, l + 20, 32);

            // state update (meaningful on lanes 0..9)
            c = fmaf(fg, c, act1 * gg);
            const float th = fmaf(2.f,
                __builtin_amdgcn_rcpf(1.f + __expf(-2.f * c)), -1.f);
            const float hN = act2 * th;

            // folded output projection: sum over lanes 0..9
            float p = (l < 10) ? (hN * wo) : 0.f;
            p += __shfl_xor(p, 1, 32);
            p += __shfl_xor(p, 2, 32);
            p += __shfl_xor(p, 4, 32);
            p += __shfl_xor(p, 8, 32);
            if (l == 0) out[t] = fmaxf(p + bo, 0.f);

            // publish h for next step (in-order LDS + single-wave barrier)
            if (l < 10) hsh[l] = hN;
            __syncthreads();
        }
    }
}

// ---------------------------------------------------------------------------
extern "C" void kernel_launch(void* const* d_in, const int* in_sizes, int n_in,
                              void* d_out, int out_size, void* d_ws, size_t ws_size,
                              hipStream_t stream) {
    const float* x     = (const float*)d_in[0];
    const float* W_ih  = (const float*)d_in[1];
    const float* W_hh  = (const float*)d_in[2];
    const float* b_ih  = (const float*)d_in[3];
    const float* b_hh  = (const float*)d_in[4];
    const float* W_out = (const float*)d_in[5];
    const float* b_out = (const float*)d_in[6];
    float* out   = (float*)d_out;
    float* xproj = (float*)d_ws;            // T*40 floats = ~21 MB

    const int T      = in_sizes[0] / 30;    // 131072
    const int Mtiles = (T + 15) / 16;       // 8192
    const int wavesPerBlock = 8;            // 256 threads
    const int blocks = (Mtiles + wavesPerBlock - 1) / wavesPerBlock;

    xproj_wmma<<<blocks, 256, 0, stream>>>(x, W_ih, b_ih, b_hh, xproj, Mtiles);
    lstm_seq<<<1, 32, 0, stream>>>(xproj, W_hh, W_out, b_out, out, T);
}